// UnivNetLVCResidualBlock_4260607558287
// MI455X (gfx1250) — compile-verified
//
#include <hip/hip_runtime.h>
#include <hip/hip_bf16.h>
#include <math.h>

typedef float v2f __attribute__((ext_vector_type(2)));
typedef float v8f __attribute__((ext_vector_type(8)));

#define HIDC   64
#define OC     128
#define KS     3
#define LTOT   16384
#define HOP    256
#define KLN    64
#define KDIM   192          // C*Ks
#define SLOPE  0.2f

#define H_STRIDE  280       // input window row stride (264 used, 16B-aligned rows)
#define W_STRIDE  196       // conv-weight row stride: 16B-aligned, bank-perfect b64 frags
#define XC_STRIDE 276       // conv-output row stride: lane-halves hit disjoint bank ranges
#define A_STRIDE  196       // LVC A row stride: 16B-aligned, bank-perfect b64 frags
#define T_STRIDE  196       // transpose tile row stride (16B-aligned for async b128 store)

// ---- CDNA5 async global<->LDS path (guarded; falls back to scalar copies) ----
#if defined(__has_builtin)
#if __has_builtin(__builtin_amdgcn_global_load_async_to_lds_b128) && \
    __has_builtin(__builtin_amdgcn_global_load_async_to_lds_b32)  && \
    __has_builtin(__builtin_amdgcn_global_store_async_from_lds_b128) && \
    __has_builtin(__builtin_amdgcn_s_wait_asynccnt)
#define USE_ASYNC 1
#endif
#endif
#ifndef USE_ASYNC
#define USE_ASYNC 0
#endif

#if USE_ASYNC
// pointee types exactly as reported by clang's diagnostics:
//   b32  : '__device__ int *'            /  '__shared__ int *'
//   b128 : 'vector_size(16) int __device__ *' / shared equivalent
typedef int vsi4 __attribute__((vector_size(16)));
typedef __attribute__((address_space(1))) vsi4 g_v4i;
typedef __attribute__((address_space(3))) vsi4 l_v4i;
typedef __attribute__((address_space(1))) int  g_i32;
typedef __attribute__((address_space(3))) int  l_i32;

__device__ __forceinline__ void async_cp128(const void* g, void* l) {
    __builtin_amdgcn_global_load_async_to_lds_b128((g_v4i*)g, (l_v4i*)l, 0, 0);
}
__device__ __forceinline__ void async_cp32(const void* g, void* l) {
    __builtin_amdgcn_global_load_async_to_lds_b32((g_i32*)g, (l_i32*)l, 0, 0);
}
__device__ __forceinline__ void async_st128(void* g, const void* l) {
    __builtin_amdgcn_global_store_async_from_lds_b128((g_v4i*)g, (l_v4i*)l, 0, 0);
}
__device__ __forceinline__ void async_wait0() { __builtin_amdgcn_s_wait_asynccnt(0); }
#endif

__device__ __forceinline__ v8f wmma4(v2f a, v2f b, v8f c) {
    return __builtin_amdgcn_wmma_f32_16x16x4_f32(false, a, false, b, (short)0, c, false, false);
}
__device__ __forceinline__ float lrelu(float x)    { return x >= 0.f ? x : SLOPE * x; }
__device__ __forceinline__ float sigmoidf(float x) { return 1.f / (1.f + __expf(-x)); }
__device__ __forceinline__ float fast_tanh(float x){ return 1.f - 2.f / (1.f + __expf(2.f * x)); }

// ---------------------------------------------------------------------------
// Kernel 1: transpose kernel (B,C,O,Ks,KL) -> A_t[b][kl][o][kk], kk = k*64+c
// One block per (o, b). Async engine does both the permuted gather into LDS
// (per-lane b32 dests) and the coalesced b128 store back out.
// ---------------------------------------------------------------------------
__global__ __launch_bounds__(256) void lvc_kernel_transpose(
        const float* __restrict__ kern, float* __restrict__ a_t) {
    __shared__ float tile[64 * T_STRIDE];
    const int o = blockIdx.x;      // 0..127
    const int b = blockIdx.y;      // 0..7
    const int tid = threadIdx.x;

#if USE_ASYNC
    #pragma unroll 4
    for (int it = 0; it < 48; ++it) {
        int i  = it * 256 + tid;          // 0..12287
        int kl = i & 63;
        int ck = i >> 6;
        int c  = ck / 3;
        int k  = ck - c * 3;
        async_cp32(&kern[((((size_t)b * HIDC + c) * OC + o) * KS + k) * KLN + kl],
                   &tile[kl * T_STRIDE + k * 64 + c]);
    }
    async_wait0();
    __syncthreads();
    #pragma unroll 4
    for (int it = 0; it < 12; ++it) {
        int q  = it * 256 + tid;          // 0..3071 (chunks of 4 floats)
        int kc = q % 48;
        int kl = q / 48;
        async_st128(&a_t[(((size_t)b * KLN + kl) * OC + o) * KDIM + kc * 4],
                    &tile[kl * T_STRIDE + kc * 4]);
    }
    async_wait0();
#else
    #pragma unroll 4
    for (int it = 0; it < 48; ++it) {
        int i  = it * 256 + tid;
        int kl = i & 63;
        int ck = i >> 6;
        int c  = ck / 3;
        int k  = ck - c * 3;
        tile[kl * T_STRIDE + k * 64 + c] =
            kern[((((size_t)b * HIDC + c) * OC + o) * KS + k) * KLN + kl];
    }
    __syncthreads();
    #pragma unroll 4
    for (int it = 0; it < 48; ++it) {
        int j  = it * 256 + tid;
        int kk = j % 192;
        int kl = j / 192;
        a_t[(((size_t)b * KLN + kl) * OC + o) * KDIM + kk] = tile[kl * T_STRIDE + kk];
    }
#endif
}

// ---------------------------------------------------------------------------
// Kernel 2: fused lrelu -> dilated conv -> lrelu -> LVC -> gate -> residual.
// One 256-thread (8-wave) workgroup per (kl, b) chunk of 256 columns.
// ---------------------------------------------------------------------------
__global__ __launch_bounds__(256) void lvc_block_main(
        const float* __restrict__ hs,     // (8,64,16384)
        const float* __restrict__ a_t,    // (8,64,128,192) transposed LVC kernel
        const float* __restrict__ bias,   // (8,128,64)
        const float* __restrict__ cw,     // (64,64,3) = [o][kk], kk=c*3+k
        const float* __restrict__ cb,     // (64,)
        float* __restrict__ out)          // (8,64,16384)
{
    __shared__ float s_h [HIDC * H_STRIDE];   //  71.7 KB lrelu'd input window
    __shared__ float s_w [HIDC * W_STRIDE];   //  50.2 KB conv weights
    __shared__ float s_xc[HIDC * XC_STRIDE];  //  70.7 KB conv output window
    __shared__ float s_a [OC   * A_STRIDE];   // 100.4 KB LVC kernel slice
    __shared__ float s_bias[OC];
    __shared__ float s_cb [HIDC];

    const int kl   = blockIdx.x;              // 0..63
    const int b    = blockIdx.y;              // 0..7
    const int tid  = threadIdx.x;
    const int lane = tid & 31;
    const int wv   = tid >> 5;                // wave 0..7 (wave32)
    const int l16  = lane & 15;
    const int hi   = lane >> 4;               // 0: K=+0,+1 / M=r ; 1: K=+2,+3 / M=8+r
    const int base = kl * HOP;
    const float* aslice = a_t + ((size_t)(b * KLN + kl)) * (OC * KDIM);

    // ---- stage 0: LDS fills -------------------------------------------------
#if USE_ASYNC
    // pure copies go through the async-to-LDS engine (no VGPR staging)
    for (int q = tid; q < HIDC * 48; q += 256) {          // s_w: 64 rows x 48 b128
        int o = q / 48, ch = q - o * 48;
        async_cp128(cw + o * KDIM + ch * 4, &s_w[o * W_STRIDE + ch * 4]);
    }
    for (int q = tid; q < OC * 48; q += 256) {            // s_a: 128 rows x 48 b128
        int o = q / 48, ch = q - o * 48;
        async_cp128(aslice + o * KDIM + ch * 4, &s_a[o * A_STRIDE + ch * 4]);
    }
#else
    for (int i = tid; i < HIDC * KDIM; i += 256) {
        int o = i / KDIM, kk = i - o * KDIM;
        s_w[o * W_STRIDE + kk] = cw[i];
    }
    for (int i = tid; i < OC * KDIM; i += 256) {
        int o = i / KDIM, kk = i - o * KDIM;
        s_a[o * A_STRIDE + kk] = aslice[i];
    }
#endif
    // input window needs lrelu + boundary zero-fill -> scalar path
    for (int i = tid; i < HIDC * H_STRIDE; i += 256) {
        int c  = i / H_STRIDE;
        int ht = i - c * H_STRIDE;
        int gc = base - 4 + ht;
        float v = 0.f;
        if (ht < 264 && gc >= 0 && gc < LTOT)
            v = lrelu(hs[((size_t)b * HIDC + c) * LTOT + gc]);
        s_h[i] = v;
    }
    for (int i = tid; i < OC;   i += 256) s_bias[i] = bias[((size_t)b * OC + i) * KLN + kl];
    for (int i = tid; i < HIDC; i += 256) s_cb[i]   = cb[i];
#if USE_ASYNC
    async_wait0();
#endif
    __syncthreads();

    // ---- stage 1: dilated conv as WMMA GEMM: M=64, N=272, K=192 ------------
    // xc[m][t] = lrelu( sum_kk w[m][kk] * h[c][t + 3*k] + cb[m] ),  kk = c*3+k
    // M-tile pairs share the im2col B fragment; two independent WMMA chains.
    for (int P = wv; P < 34; P += 8) {        // 17 N-tiles x 2 M-tile-pairs
        const int nt  = P >> 1;
        const int mt0 = (P & 1) * 2;          // 0 or 2
        const int t   = nt * 16 + l16;
        v8f acc0 = {0.f,0.f,0.f,0.f,0.f,0.f,0.f,0.f};
        v8f acc1 = {0.f,0.f,0.f,0.f,0.f,0.f,0.f,0.f};
        #pragma unroll
        for (int st = 0; st < 48; ++st) {
            const int kk0 = st * 4 + 2 * hi;
            v2f a0 = *(const v2f*)&s_w[( mt0      * 16 + l16) * W_STRIDE + kk0];
            v2f a1 = *(const v2f*)&s_w[((mt0 + 1) * 16 + l16) * W_STRIDE + kk0];
            const int kkA = kk0,     cA = kkA / 3, kA = kkA - cA * 3;
            const int kkB = kk0 + 1, cB = kkB / 3, kB = kkB - cB * 3;
            v2f bb = { s_h[cA * H_STRIDE + t + 3 * kA],
                       s_h[cB * H_STRIDE + t + 3 * kB] };
            acc0 = wmma4(a0, bb, acc0);
            acc1 = wmma4(a1, bb, acc1);
        }
        #pragma unroll
        for (int r = 0; r < 8; ++r) {
            int m0 = mt0 * 16 + hi * 8 + r;
            s_xc[ m0       * XC_STRIDE + t] = lrelu(acc0[r] + s_cb[m0]);
            s_xc[(m0 + 16) * XC_STRIDE + t] = lrelu(acc1[r] + s_cb[m0 + 16]);
        }
    }
    __syncthreads();

    // ---- stage 2: LVC GEMM M=128, N=256, K=192 + gating + residual ----------
    // Wave owns M-tile pair (p, p+4) => channels [16p,16p+16) and +64, so the
    // sigmoid(x_a)*tanh(x_b) gate fuses into the epilogue with no LDS bounce.
    const int p     = wv & 3;
    const int nbase = (wv >> 2) * 8;
    const int oA    = p * 16;                 // sigmoid half rows
    const int oB    = 64 + p * 16;            // tanh half rows

    for (int nb = 0; nb < 4; ++nb) {          // 2 N-tiles per iteration
        const int nt0 = nbase + nb * 2;
        const int s0  = nt0 * 16 + l16;
        const int s1  = s0 + 16;
        v8f accA0 = {0.f,0.f,0.f,0.f,0.f,0.f,0.f,0.f};
        v8f accA1 = {0.f,0.f,0.f,0.f,0.f,0.f,0.f,0.f};
        v8f accB0 = {0.f,0.f,0.f,0.f,0.f,0.f,0.f,0.f};
        v8f accB1 = {0.f,0.f,0.f,0.f,0.f,0.f,0.f,0.f};
        #pragma unroll
        for (int st = 0; st < 48; ++st) {
            const int kk0 = st * 4 + 2 * hi;
            v2f aA = *(const v2f*)&s_a[(oA + l16) * A_STRIDE + kk0];   // b64, conflict-free
            v2f aB = *(const v2f*)&s_a[(oB + l16) * A_STRIDE + kk0];
            const int c0 = kk0 & 63,       t0 = kk0 >> 6;              // kk = k*64+c
            const int c1 = (kk0 + 1) & 63, t1 = (kk0 + 1) >> 6;
            v2f b0 = { s_xc[c0 * XC_STRIDE + s0 + t0],
                       s_xc[c1 * XC_STRIDE + s0 + t1] };
            v2f b1 = { s_xc[c0 * XC_STRIDE + s1 + t0],
                       s_xc[c1 * XC_STRIDE + s1 + t1] };
            accA0 = wmma4(aA, b0, accA0);   // B frags reused by both M-tiles
            accB0 = wmma4(aB, b0, accB0);
            accA1 = wmma4(aA, b1, accA1);
            accB1 = wmma4(aB, b1, accB1);
        }
        #pragma unroll
        for (int n = 0; n < 2; ++n) {
            const v8f& aAcc = n ? accA1 : accA0;
            const v8f& bAcc = n ? accB1 : accB0;
            const int  col  = base + (nt0 + n) * 16 + l16;
            #pragma unroll
            for (int r = 0; r < 8; ++r) {
                int ch = oA + hi * 8 + r;                       // 0..63
                float xa = aAcc[r] + s_bias[ch];
                float xb = bAcc[r] + s_bias[ch + 64];
                float g  = sigmoidf(xa) * fast_tanh(xb);
                size_t oi = ((size_t)b * HIDC + ch) * LTOT + col;
                out[oi] = hs[oi] + g;
            }
        }
    }
}

// ---------------------------------------------------------------------------
extern "C" void kernel_launch(void* const* d_in, const int* in_sizes, int n_in,
                              void* d_out, int out_size, void* d_ws, size_t ws_size,
                              hipStream_t stream) {
    const float* hs   = (const float*)d_in[0];   // (8,64,16384)
    const float* kern = (const float*)d_in[1];   // (8,64,128,3,64)
    const float* bias = (const float*)d_in[2];   // (8,128,64)
    const float* cw   = (const float*)d_in[3];   // (64,64,3)
    const float* cb   = (const float*)d_in[4];   // (64,)
    float*       out  = (float*)d_out;
    float*       a_t  = (float*)d_ws;            // 8*64*128*192*4 = 50.3 MB

    lvc_kernel_transpose<<<dim3(OC, 8), 256, 0, stream>>>(kern, a_t);
    lvc_block_main<<<dim3(KLN, 8), 256, 0, stream>>>(hs, a_t, bias, cw, cb, out);
}